// relative_loss_9500467659018
// MI455X (gfx1250) — compile-verified
//
#include <hip/hip_runtime.h>
#include <hip/hip_bf16.h>
#include <math.h>

// Problem constants (from reference)
#define B_ROWS 512
#define K_DIM  2048
#define N_DIM  312
#define N_PAD  320   // 20 tiles of 16

typedef float v2f __attribute__((ext_vector_type(2)));
typedef float v8f __attribute__((ext_vector_type(8)));

// ---------------------------------------------------------------------------
// Kernel 1: fused fp32 WMMA GEMM (pre = x_f @ W) + per-row log-softmax stats.
// Grid: 32 blocks (one per 16-row tile of x_f). Block: 128 threads = 4 waves.
// Wave w computes columns [w*80, w*80+80) as 5 WMMA 16x16 tiles over K=2048.
// Then the block stages the 16x320 tile in LDS and 16 threads compute
//   S[row] = sum_d pre[row,d] - D*(max_d pre + log sum_d exp(pre - max))
// with bias b added on the fly (b is part of pre_attr in the reference).
// ---------------------------------------------------------------------------
__global__ __launch_bounds__(128) void gemm_rowstat_kernel(
    const float* __restrict__ xf,   // [512, 2048]
    const float* __restrict__ W,    // [2048, 312]
    const float* __restrict__ bias, // [312]
    float* __restrict__ Srow)       // [512] workspace
{
  __shared__ float tile[16 * N_PAD];   // 20 KB

  const int m    = blockIdx.x;             // row-tile index 0..31
  const int lane = threadIdx.x & 31;
  const int wave = threadIdx.x >> 5;       // 0..3
  const int half = lane >> 4;              // 0: K+{0,1}, 1: K+{2,3}
  const int l15  = lane & 15;

  v8f acc[5] = {};                         // 5 N-tiles of 16x16 f32 accum

  // A fragment source: row = m*16 + l15, contiguous pair of K values.
  const int arow = m * 16 + l15;
  const float* aptr = xf + (size_t)arow * K_DIM + 2 * half;

  // B fragment columns for each of this wave's 5 tiles (clamped for padding;
  // padded columns are excluded from the reduction below, so garbage is OK
  // and the clamp keeps all loads in-bounds).
  int cols[5];
#pragma unroll
  for (int t = 0; t < 5; ++t) {
    int c = wave * 80 + t * 16 + l15;
    cols[t] = (c < N_DIM) ? c : (N_DIM - 1);
  }

  for (int k = 0; k < K_DIM; k += 4) {
    // A: 16x4 tile, 2 consecutive floats per lane (8B aligned).
    v2f a;
    a.x = aptr[k];
    a.y = aptr[k + 1];
    // prefetch the A stream ahead (global_prefetch_b8 on gfx1250)
    if (k + 256 < K_DIM) __builtin_prefetch(aptr + k + 256, 0, 1);

    // B: 4x16 tile per N-tile; lanes 0-15 rows {k,k+1}, lanes 16-31 {k+2,k+3}
    const float* brow0 = W + (size_t)(k + 2 * half) * N_DIM;
    const float* brow1 = brow0 + N_DIM;
#pragma unroll
    for (int t = 0; t < 5; ++t) {
      v2f b;
      b.x = brow0[cols[t]];
      b.y = brow1[cols[t]];
      // D = A(16x4, f32) * B(4x16, f32) + C ; native fp32 matrix pipe
      acc[t] = __builtin_amdgcn_wmma_f32_16x16x4_f32(
          /*neg_a=*/false, a, /*neg_b=*/false, b,
          /*c_mod=*/(short)0, acc[t], /*reuse_a=*/false, /*reuse_b=*/false);
    }
  }

  // Spill accumulators to LDS: C/D layout is VGPR r -> M = r + 8*half,
  // lane&15 -> N within tile.
#pragma unroll
  for (int t = 0; t < 5; ++t) {
    int col = wave * 80 + t * 16 + l15;
#pragma unroll
    for (int r = 0; r < 8; ++r) {
      int row = r + 8 * half;
      tile[row * N_PAD + col] = acc[t][r];
    }
  }
  __syncthreads();

  // Row statistics: one thread per row (16 rows), columns 0..311 only.
  if (threadIdx.x < 16) {
    const int row = threadIdx.x;
    const float* tr = &tile[row * N_PAD];
    float mx = -INFINITY;
    float sum = 0.0f;
    for (int c = 0; c < N_DIM; ++c) {
      float v = tr[c] + bias[c];
      sum += v;
      mx = fmaxf(mx, v);
    }
    float se = 0.0f;
    for (int c = 0; c < N_DIM; ++c) {
      float v = tr[c] + bias[c];
      se += expf(v - mx);
    }
    Srow[m * 16 + row] = sum - (float)N_DIM * (mx + logf(se));
  }
}

// ---------------------------------------------------------------------------
// Kernel 2: masked pairwise reduction.
//   loss  = sum_{i<j, lab_i==lab_j} (S_j - S_i)
//   count = number of such pairs
//   out   = count > 0 ? loss/count : loss
// Single block of 512 threads; wave32 shuffle + LDS tree reduction.
// ---------------------------------------------------------------------------
__global__ __launch_bounds__(512) void pair_reduce_kernel(
    const float* __restrict__ Srow,   // [512]
    const int* __restrict__ labels,   // [512]
    float* __restrict__ out)          // [1]
{
  __shared__ float Ssh[B_ROWS];
  __shared__ int   Lsh[B_ROWS];
  __shared__ float redL[16];
  __shared__ int   redC[16];

  const int j = threadIdx.x;
  Ssh[j] = Srow[j];
  Lsh[j] = labels[j];
  __syncthreads();

  const float Sj = Ssh[j];
  const int   Lj = Lsh[j];
  float loss = 0.0f;
  int   cnt  = 0;
  for (int i = 0; i < j; ++i) {
    if (Lsh[i] == Lj) {
      loss += Sj - Ssh[i];
      ++cnt;
    }
  }

  // wave32 reduction
#pragma unroll
  for (int off = 16; off > 0; off >>= 1) {
    loss += __shfl_down(loss, off, 32);
    cnt  += __shfl_down(cnt,  off, 32);
  }
  if ((threadIdx.x & 31) == 0) {
    redL[threadIdx.x >> 5] = loss;
    redC[threadIdx.x >> 5] = cnt;
  }
  __syncthreads();

  if (threadIdx.x == 0) {
    float L = 0.0f;
    int   C = 0;
#pragma unroll
    for (int w = 0; w < 16; ++w) { L += redL[w]; C += redC[w]; }
    out[0] = (C > 0) ? (L / (float)C) : L;
  }
}

// ---------------------------------------------------------------------------
extern "C" void kernel_launch(void* const* d_in, const int* in_sizes, int n_in,
                              void* d_out, int out_size, void* d_ws, size_t ws_size,
                              hipStream_t stream) {
  const float* xf      = (const float*)d_in[0];  // [512,2048]
  const float* W       = (const float*)d_in[1];  // [2048,312]
  const float* bias    = (const float*)d_in[2];  // [312]
  // d_in[3] = seen_att: cancels exactly for same-label pairs; unused.
  const int*   labels  = (const int*)d_in[4];    // [512]
  float*       out     = (float*)d_out;          // [1]
  float*       Srow    = (float*)d_ws;           // 512 floats of scratch

  gemm_rowstat_kernel<<<dim3(B_ROWS / 16), dim3(128), 0, stream>>>(xf, W, bias, Srow);
  pair_reduce_kernel<<<dim3(1), dim3(512), 0, stream>>>(Srow, labels, out);
}